// TanhAttention_52699248722206
// MI455X (gfx1250) — compile-verified
//
#include <hip/hip_runtime.h>

#define B_ 4
#define L_ 512
#define D_ 256

typedef __attribute__((ext_vector_type(2))) float v2f;
typedef __attribute__((ext_vector_type(8))) float v8f;

// tanh(x) = (e^(2x)-1)/(e^(2x)+1), via hardware v_exp_f32 (base 2) + v_rcp_f32.
// Clamp keeps exp2 finite; for |x|>10, tanh is +-1 to f32 precision anyway.
__device__ __forceinline__ float fast_tanh(float x) {
    x = fminf(10.0f, fmaxf(-10.0f, x));
    float e = __builtin_amdgcn_exp2f(x * 2.8853900817779268f); // 2*log2(e)
    return (e - 1.0f) * __builtin_amdgcn_rcpf(e + 1.0f);
}

// ---------------------------------------------------------------------------
// Stage 1: scores[b,i,j] = sum_d tanh(H[b,i,d]+H[b,j,d]) * w[d] + bias
// One block = 16x16 (i,j) tile; H tiles + w staged in LDS; one score/thread.
// ---------------------------------------------------------------------------
__global__ __launch_bounds__(256) void ta_scores_kernel(
    const float* __restrict__ H, const float* __restrict__ w,
    const float* __restrict__ bias, float* __restrict__ scores)
{
    __shared__ float His[16][D_ + 4];   // +4 pad: bank-conflict-free, 16B aligned
    __shared__ float Hjs[16][D_ + 4];
    __shared__ float ws[D_];

    const int b   = blockIdx.z;
    const int i0  = blockIdx.y * 16;
    const int j0  = blockIdx.x * 16;
    const int tid = threadIdx.x;

    const float* Hb = H + (size_t)b * L_ * D_;

    // cooperative tile load, coalesced float4s
    for (int idx = tid; idx < 16 * (D_ / 4); idx += 256) {
        const int row = idx / (D_ / 4);
        const int c4  = (idx % (D_ / 4)) * 4;
        float4 vi = *(const float4*)(Hb + (size_t)(i0 + row) * D_ + c4);
        float4 vj = *(const float4*)(Hb + (size_t)(j0 + row) * D_ + c4);
        *(float4*)&His[row][c4] = vi;
        *(float4*)&Hjs[row][c4] = vj;
    }
    ws[tid] = w[tid];   // D_ == 256 == blockDim
    __syncthreads();

    const int i = tid >> 4;
    const int j = tid & 15;
    const float* hi = &His[i][0];
    const float* hj = &Hjs[j][0];

    float s = 0.0f;
    #pragma unroll 4
    for (int d = 0; d < D_; d += 4) {
        float4 a  = *(const float4*)(hi + d);
        float4 c  = *(const float4*)(hj + d);
        float4 wv = *(const float4*)(ws + d);
        s += fast_tanh(a.x + c.x) * wv.x;
        s += fast_tanh(a.y + c.y) * wv.y;
        s += fast_tanh(a.z + c.z) * wv.z;
        s += fast_tanh(a.w + c.w) * wv.w;
    }
    scores[((size_t)b * L_ + (i0 + i)) * L_ + (j0 + j)] = s + bias[0];
}

// ---------------------------------------------------------------------------
// Stage 2: in-place row softmax over j (L_=512 elems), one block per (b,i).
// ---------------------------------------------------------------------------
__global__ __launch_bounds__(256) void ta_softmax_kernel(float* __restrict__ alpha)
{
    __shared__ float red[256];
    const int b = blockIdx.y;
    const int i = blockIdx.x;
    float* row = alpha + ((size_t)b * L_ + i) * L_;
    const int tid = threadIdx.x;

    float s0 = row[tid];
    float s1 = row[tid + 256];

    red[tid] = fmaxf(s0, s1);
    __syncthreads();
    for (int off = 128; off > 0; off >>= 1) {
        if (tid < off) red[tid] = fmaxf(red[tid], red[tid + off]);
        __syncthreads();
    }
    const float m = red[0];
    __syncthreads();

    const float k = 1.4426950408889634f; // log2(e)
    float e0 = __builtin_amdgcn_exp2f((s0 - m) * k);
    float e1 = __builtin_amdgcn_exp2f((s1 - m) * k);

    red[tid] = e0 + e1;
    __syncthreads();
    for (int off = 128; off > 0; off >>= 1) {
        if (tid < off) red[tid] += red[tid + off];
        __syncthreads();
    }
    const float inv = 1.0f / red[0];
    row[tid]       = e0 * inv;
    row[tid + 256] = e1 * inv;
}

// ---------------------------------------------------------------------------
// Stage 3: r[b,i,d] = sum_j alpha[b,i,j] * H[b,j,d]
// Batched GEMM (M=512, N=256, K=512) via V_WMMA_F32_16X16X4_F32.
// One wave (blockDim=32, EXEC all-ones as WMMA requires) per 16x16 r-tile.
//
// ISA fragment layouts (wave32):
//   A 16x4 f32 (2 VGPRs): lane l -> M=l%16; VGPR v -> K = v + 2*(l/16)
//   B  4x16 f32 (2 VGPRs): lane l -> N=l%16; VGPR v -> K = v + 2*(l/16)
//   C/D 16x16 f32 (8 VGPRs): lane l, VGPR v -> M = v + 8*(l/16), N = l%16
// ---------------------------------------------------------------------------
__global__ __launch_bounds__(32) void ta_r_wmma_kernel(
    const float* __restrict__ alpha, const float* __restrict__ H,
    float* __restrict__ r)
{
    const int MT = L_ / 16;   // 32 tiles in i
    const int NT = D_ / 16;   // 16 tiles in d
    int t = blockIdx.x;
    const int b  = t / (MT * NT);
    t %= (MT * NT);
    const int mi = t / NT;
    const int ni = t % NT;

    const int lane = threadIdx.x;  // 0..31
    const int half = lane >> 4;    // 0 or 1
    const int lid  = lane & 15;

    const float* A  = alpha + (size_t)b * L_ * L_;  // [L][L], row=i, col=j(K)
    const float* Bm = H     + (size_t)b * L_ * D_;  // [L][D], row=j(K), col=d(N)
    float*       Rb = r     + (size_t)b * L_ * D_;

    const int i = mi * 16 + lid;   // A-row held by this lane
    const int n = ni * 16 + lid;   // B-col held by this lane

    v8f c = {};
    for (int k0 = 0; k0 < L_; k0 += 4) {
        const int ka = k0 + 2 * half;
        v2f av, bv;
        av.x = A[(size_t)i * L_ + ka];        // consecutive j -> b64 load
        av.y = A[(size_t)i * L_ + ka + 1];
        bv.x = Bm[(size_t)ka       * D_ + n];
        bv.y = Bm[(size_t)(ka + 1) * D_ + n];
        c = __builtin_amdgcn_wmma_f32_16x16x4_f32(
                /*neg_a=*/false, av, /*neg_b=*/false, bv,
                /*c_mod=*/(short)0, c, /*reuse_a=*/false, /*reuse_b=*/false);
    }

    #pragma unroll
    for (int v = 0; v < 8; ++v) {
        const int mrow = mi * 16 + v + 8 * half;
        Rb[(size_t)mrow * D_ + (ni * 16 + lid)] = c[v];
    }
}

// ---------------------------------------------------------------------------
extern "C" void kernel_launch(void* const* d_in, const int* in_sizes, int n_in,
                              void* d_out, int out_size, void* d_ws, size_t ws_size,
                              hipStream_t stream) {
    (void)in_sizes; (void)n_in; (void)out_size; (void)d_ws; (void)ws_size;

    const float* H    = (const float*)d_in[0];
    const float* w    = (const float*)d_in[1];
    const float* bias = (const float*)d_in[2];

    // d_out = [ r : B*L*D floats | alpha : B*L*L floats ]
    float* r     = (float*)d_out;
    float* alpha = (float*)d_out + (size_t)B_ * L_ * D_;

    // Stage 1: scores -> alpha region
    dim3 g1(L_ / 16, L_ / 16, B_);
    ta_scores_kernel<<<g1, 256, 0, stream>>>(H, w, bias, alpha);

    // Stage 2: softmax in place
    dim3 g2(L_, B_);
    ta_softmax_kernel<<<g2, 256, 0, stream>>>(alpha);

    // Stage 3: r = alpha @ H via WMMA (one wave per 16x16 tile)
    const int ntiles = B_ * (L_ / 16) * (D_ / 16);  // 2048
    ta_r_wmma_kernel<<<ntiles, 32, 0, stream>>>(alpha, H, r);
}